// ScreenedPoissonLayer_85023172591716
// MI455X (gfx1250) — compile-verified
//
#include <hip/hip_runtime.h>
#include <hip/hip_bf16.h>
#include <stdint.h>

// ---------------------------------------------------------------------------
// Screened Poisson via Jacobi-preconditioned CG, 50 fixed iterations.
// B=4, C=1, H=W=1024, fp32. Bandwidth-bound stencil; CG state (~96MB) fits in
// the MI455X 192MB L2, so the whole solve is L2-resident after the first sweep.
// CDNA5-specific path: async global->LDS halo-tile staging (ASYNCcnt) in the
// stencil kernels; wave32 shuffle reductions for the CG dot products.
// ---------------------------------------------------------------------------

#define TX 32
#define TY 8
#define NT 256                 // threads per block (8 wave32)
#define NWAVE (NT / 32)
#define SU_S (TX + 2)          // u-tile row stride (halo on all 4 sides)
#define SW_S (TX + 1)          // wg-tile row stride (halo left/top only)
#define U_ELEMS ((TY + 2) * SU_S)
#define W_ELEMS ((TY + 1) * SW_S)
#define EPS_PC 1e-8f
#define EPS_DOT 1e-12f
#define MAX_ITER 50

// ---- CDNA5 async global->LDS copy (falls back to plain LDS store) ----------
#if defined(__has_builtin)
#if __has_builtin(__builtin_amdgcn_global_load_async_to_lds_b32)
#define HAVE_ASYNC_LDS 1
#endif
#endif

typedef __attribute__((address_space(1))) int g_int;
typedef __attribute__((address_space(3))) int l_int;

__device__ __forceinline__ void async_cp_f32(const float* g, float* l) {
#ifdef HAVE_ASYNC_LDS
  __builtin_amdgcn_global_load_async_to_lds_b32(
      (g_int*)g, (l_int*)l, /*imm offset*/ 0, /*cpol*/ 0);
#else
  *l = *g;
#endif
}

__device__ __forceinline__ void async_wait_all() {
#ifdef HAVE_ASYNC_LDS
#if __has_builtin(__builtin_amdgcn_s_wait_asynccnt)
  __builtin_amdgcn_s_wait_asynccnt(0);
#else
  asm volatile("s_wait_asynccnt 0" ::: "memory");
#endif
#endif
}

__device__ __forceinline__ int iclamp(int v, int lo, int hi) {
  return v < lo ? lo : (v > hi ? hi : v);
}

// Stage u-tile (4-sided halo) and wg-tile (left/top halo) into LDS.
// Halo addresses are clamped into the domain; out-of-domain values are killed
// by zero boundary weights in the stencil, so clamped data is harmless.
__device__ __forceinline__ void load_halo_tiles(const float* __restrict__ gU,
                                                const float* __restrict__ gW,
                                                size_t base, int by0, int bx0,
                                                int H, int W, float* sU,
                                                float* sW, int tid) {
  for (int idx = tid; idx < U_ELEMS; idx += NT) {
    int rr = idx / SU_S, cc = idx % SU_S;
    int gy = iclamp(by0 - 1 + rr, 0, H - 1);
    int gx = iclamp(bx0 - 1 + cc, 0, W - 1);
    async_cp_f32(&gU[base + (size_t)gy * W + gx], &sU[idx]);
  }
  for (int idx = tid; idx < W_ELEMS; idx += NT) {
    int rr = idx / SW_S, cc = idx % SW_S;
    int gy = iclamp(by0 - 1 + rr, 0, H - 1);
    int gx = iclamp(bx0 - 1 + cc, 0, W - 1);
    async_cp_f32(&gW[base + (size_t)gy * W + gx], &sW[idx]);
  }
  async_wait_all();
  __syncthreads();
}

// wave32 shuffle reduction (fixed order -> deterministic)
__device__ __forceinline__ float wave_reduce(float v) {
#pragma unroll
  for (int off = 16; off > 0; off >>= 1) v += __shfl_down(v, off, 32);
  return v;
}

// block reduction: shuffle within each wave, 8 partials through LDS, final
// shuffle in wave 0. One barrier total.
__device__ __forceinline__ void block_reduce_store(float v, float* red, int tid,
                                                   float* slot) {
  const float wv = wave_reduce(v);
  if ((tid & 31) == 0) red[tid >> 5] = wv;
  __syncthreads();
  if (tid < 32) {
    float s = (tid < NWAVE) ? red[tid] : 0.0f;
#pragma unroll
    for (int off = NWAVE / 2; off > 0; off >>= 1) s += __shfl_down(s, off, 32);
    if (tid == 0) *slot = s;
  }
}

// ---------------------------------------------------------------------------
// Setup: b (stored in r), c = lam + mu*M, inv = 1/(Mdiag+eps), p = z = b*inv,
// x = 0, partials of rz0 = b.(b*inv).
// ---------------------------------------------------------------------------
__global__ void k_init(const float* __restrict__ Ehat,
                       const float* __restrict__ DL,
                       const float* __restrict__ M,
                       const float* __restrict__ lam,
                       const float* __restrict__ mu,
                       const float* __restrict__ wg, float* __restrict__ r,
                       float* __restrict__ p, float* __restrict__ cvec,
                       float* __restrict__ invd, float* __restrict__ x,
                       float* __restrict__ partials, int H, int W, int nblk) {
  __shared__ float sU[U_ELEMS];
  __shared__ float sW[W_ELEMS];
  __shared__ float red[NWAVE];
  const int tid = threadIdx.y * TX + threadIdx.x;
  const int b = blockIdx.z;
  const int by0 = blockIdx.y * TY, bx0 = blockIdx.x * TX;
  const size_t base = (size_t)b * H * W;

  load_halo_tiles(Ehat, wg, base, by0, bx0, H, W, sU, sW, tid);

  const int i = by0 + threadIdx.y, j = bx0 + threadIdx.x;
  const size_t g = base + (size_t)i * W + j;

  const float u0 = sU[(threadIdx.y + 1) * SU_S + threadIdx.x + 1];
  const float uE = sU[(threadIdx.y + 1) * SU_S + threadIdx.x + 2];
  const float uW = sU[(threadIdx.y + 1) * SU_S + threadIdx.x];
  const float uS = sU[(threadIdx.y + 2) * SU_S + threadIdx.x + 1];
  const float uN = sU[threadIdx.y * SU_S + threadIdx.x + 1];
  const float w0 = sW[(threadIdx.y + 1) * SW_S + threadIdx.x + 1];
  const float wWv = sW[(threadIdx.y + 1) * SW_S + threadIdx.x];
  const float wNv = sW[threadIdx.y * SW_S + threadIdx.x + 1];

  const float wR = (j < W - 1) ? w0 : 0.f;
  const float wL = (j > 0) ? wWv : 0.f;
  const float wD = (i < H - 1) ? w0 : 0.f;
  const float wU = (i > 0) ? wNv : 0.f;

  const float lam0 = lam[g], mu0 = mu[g], M0 = M[g], DL0 = DL[g];
  const float mueff = mu0 * M0;
  const float c0 = lam0 + mueff;
  // lapW(E) = wR*(u0-uE)+wL*(u0-uW)+wD*(u0-uS)+wU*(u0-uN);  b = -lapW + lam*E + mueff*DL
  const float lapw = wR * (u0 - uE) + wL * (u0 - uW) + wD * (u0 - uS) + wU * (u0 - uN);
  const float bval = -lapw + lam0 * u0 + mueff * DL0;
  const float mdiag = c0 + wR + wL + wD + wU;
  const float iv = 1.0f / (mdiag + EPS_PC);
  const float zv = bval * iv;

  r[g] = bval;
  p[g] = zv;
  cvec[g] = c0;
  invd[g] = iv;
  x[g] = 0.0f;

  float* slot = &partials[(size_t)b * nblk + blockIdx.y * gridDim.x + blockIdx.x];
  block_reduce_store(bval * zv, red, tid, slot);
}

// ---------------------------------------------------------------------------
// Ap = c*p + sum_dirs w*(p - p_nbr); partials of p.Ap per block.
// ---------------------------------------------------------------------------
__global__ void k_apply(const float* __restrict__ p, const float* __restrict__ wg,
                        const float* __restrict__ cvec, float* __restrict__ Ap,
                        float* __restrict__ partials, int H, int W, int nblk) {
  __shared__ float sU[U_ELEMS];
  __shared__ float sW[W_ELEMS];
  __shared__ float red[NWAVE];
  const int tid = threadIdx.y * TX + threadIdx.x;
  const int b = blockIdx.z;
  const int by0 = blockIdx.y * TY, bx0 = blockIdx.x * TX;
  const size_t base = (size_t)b * H * W;

  load_halo_tiles(p, wg, base, by0, bx0, H, W, sU, sW, tid);

  const int i = by0 + threadIdx.y, j = bx0 + threadIdx.x;
  const size_t g = base + (size_t)i * W + j;

  const float u0 = sU[(threadIdx.y + 1) * SU_S + threadIdx.x + 1];
  const float uE = sU[(threadIdx.y + 1) * SU_S + threadIdx.x + 2];
  const float uW = sU[(threadIdx.y + 1) * SU_S + threadIdx.x];
  const float uS = sU[(threadIdx.y + 2) * SU_S + threadIdx.x + 1];
  const float uN = sU[threadIdx.y * SU_S + threadIdx.x + 1];
  const float w0 = sW[(threadIdx.y + 1) * SW_S + threadIdx.x + 1];
  const float wWv = sW[(threadIdx.y + 1) * SW_S + threadIdx.x];
  const float wNv = sW[threadIdx.y * SW_S + threadIdx.x + 1];

  const float wR = (j < W - 1) ? w0 : 0.f;
  const float wL = (j > 0) ? wWv : 0.f;
  const float wD = (i < H - 1) ? w0 : 0.f;
  const float wU = (i > 0) ? wNv : 0.f;

  const float apv = cvec[g] * u0 + wR * (u0 - uE) + wL * (u0 - uW) +
                    wD * (u0 - uS) + wU * (u0 - uN);
  Ap[g] = apv;

  float* slot = &partials[(size_t)b * nblk + blockIdx.y * gridDim.x + blockIdx.x];
  block_reduce_store(u0 * apv, red, tid, slot);
}

// ---------------------------------------------------------------------------
// Deterministic per-batch reduction of block partials.
// mode 0: ratio = prev/(sum+eps)  (alpha = rz/pAp)
// mode 1: ratio = sum/(prev+eps)  (beta = rz_new/rz), and sum_out <- rz_new
// ---------------------------------------------------------------------------
__global__ void k_reduce(const float* __restrict__ partials, int nblk,
                         float* __restrict__ sum_out,
                         const float* __restrict__ prev,
                         float* __restrict__ ratio_out, int mode) {
  __shared__ float red[NWAVE];
  const int b = blockIdx.x, t = threadIdx.x;
  float acc = 0.f;
  for (int i = t; i < nblk; i += NT) acc += partials[(size_t)b * nblk + i];
  const float wv = wave_reduce(acc);
  if ((t & 31) == 0) red[t >> 5] = wv;
  __syncthreads();
  if (t < 32) {
    float s = (t < NWAVE) ? red[t] : 0.0f;
#pragma unroll
    for (int off = NWAVE / 2; off > 0; off >>= 1) s += __shfl_down(s, off, 32);
    if (t == 0) {
      if (sum_out) sum_out[b] = s;
      if (ratio_out)
        ratio_out[b] = (mode == 0) ? prev[b] / (s + EPS_DOT) : s / (prev[b] + EPS_DOT);
    }
  }
}

// x += a*p; r -= a*Ap; z = r*inv; partials of r.z
__global__ void k_upd1(const float* __restrict__ p, const float* __restrict__ Ap,
                       const float* __restrict__ invd,
                       const float* __restrict__ alpha, float* __restrict__ x,
                       float* __restrict__ r, float* __restrict__ partials,
                       int n_hw, int nblk) {
  __shared__ float red[NWAVE];
  const int b = blockIdx.z, t = threadIdx.x;
  const size_t g = (size_t)b * n_hw + (size_t)blockIdx.x * NT + t;
  const float a = alpha[b];
  const float pv = p[g];
  x[g] += a * pv;
  const float rv = r[g] - a * Ap[g];
  r[g] = rv;
  const float zv = rv * invd[g];
  float* slot = &partials[(size_t)b * nblk + blockIdx.x];
  block_reduce_store(rv * zv, red, t, slot);
}

// p = r*inv + beta*p
__global__ void k_upd2(const float* __restrict__ r, const float* __restrict__ invd,
                       const float* __restrict__ beta, float* __restrict__ p,
                       int n_hw) {
  const int b = blockIdx.z;
  const size_t g = (size_t)b * n_hw + (size_t)blockIdx.x * NT + threadIdx.x;
  p[g] = r[g] * invd[g] + beta[b] * p[g];
}

// ---------------------------------------------------------------------------
extern "C" void kernel_launch(void* const* d_in, const int* in_sizes, int n_in,
                              void* d_out, int out_size, void* d_ws, size_t ws_size,
                              hipStream_t stream) {
  const int H = 1024, W = 1024;
  const int n_hw = H * W;
  const int B = in_sizes[0] / n_hw;  // = 4
  const size_t N = (size_t)B * n_hw;
  const int NBLK = (H / TY) * (W / TX);  // 4096 (== n_hw/NT for upd kernels)

  const float* Ehat = (const float*)d_in[0];
  const float* DL = (const float*)d_in[1];
  const float* M = (const float*)d_in[2];
  const float* lam = (const float*)d_in[3];
  const float* mu = (const float*)d_in[4];
  const float* wg = (const float*)d_in[5];
  float* x = (float*)d_out;

  float* ws = (float*)d_ws;
  float* r = ws;
  float* p = ws + N;
  float* Ap = ws + 2 * N;
  float* cvec = ws + 3 * N;
  float* invd = ws + 4 * N;
  float* partials = ws + 5 * N;              // B*NBLK
  float* rzA = partials + (size_t)B * NBLK;  // B
  float* rzB = rzA + B;
  float* alpha = rzB + B;
  float* beta = alpha + B;
  (void)ws_size; (void)n_in; (void)out_size;

  const dim3 stb(TX, TY, 1);
  const dim3 stg(W / TX, H / TY, B);
  const dim3 upg(n_hw / NT, 1, B);

  k_init<<<stg, stb, 0, stream>>>(Ehat, DL, M, lam, mu, wg, r, p, cvec, invd, x,
                                  partials, H, W, NBLK);
  k_reduce<<<B, NT, 0, stream>>>(partials, NBLK, rzA, nullptr, nullptr, 0);

  for (int k = 0; k < MAX_ITER; ++k) {
    float* rz_cur = (k & 1) ? rzB : rzA;
    float* rz_nxt = (k & 1) ? rzA : rzB;
    k_apply<<<stg, stb, 0, stream>>>(p, wg, cvec, Ap, partials, H, W, NBLK);
    k_reduce<<<B, NT, 0, stream>>>(partials, NBLK, nullptr, rz_cur, alpha, 0);
    k_upd1<<<upg, NT, 0, stream>>>(p, Ap, invd, alpha, x, r, partials, n_hw, NBLK);
    k_reduce<<<B, NT, 0, stream>>>(partials, NBLK, rz_nxt, rz_cur, beta, 1);
    k_upd2<<<upg, NT, 0, stream>>>(r, invd, beta, p, n_hw);
  }
}